// ODE_Vanilla_75015898792538
// MI455X (gfx1250) — compile-verified
//
#include <hip/hip_runtime.h>
#include <hip/hip_bf16.h>

typedef __attribute__((ext_vector_type(2))) float v2f;
typedef __attribute__((ext_vector_type(8))) float v8f;

#define BDIM 128
#define TDIM 256
#define IDIM 256
#define HDIM 1024
#define CDIM 10
#define LRATE 0.001f

// ---------------------------------------------------------------------------
// Fragment load helpers (ISA 7.12.2 layouts, wave32)
//  A (16x4, row-major src):  lane m=lane&15, half=lane>>4, regs hold K=2*half+{0,1}
//  B (4x16, row-major src):  lane n=lane&15, half=lane>>4, regs hold K={0,1}+2*half
//  C (16x16): reg r <-> row (r + 8*half), col = col0 + (lane&15)
// ---------------------------------------------------------------------------
__device__ __forceinline__ v2f load_a_frag(const float* __restrict__ S, int ld,
                                           int row0, int k0, int m, int half) {
    const float2 t = *(const float2*)(S + (size_t)(row0 + m) * ld + k0 + 2 * half);
    v2f a; a.x = t.x; a.y = t.y;
    return a;
}

__device__ __forceinline__ v2f load_b_frag(const float* __restrict__ S, int ld,
                                           int k0, int col0, int n, int half) {
    v2f b;
    b.x = S[(size_t)(k0 + 2 * half + 0) * ld + col0 + n];
    b.y = S[(size_t)(k0 + 2 * half + 1) * ld + col0 + n];
    return b;
}

// ---------------------------------------------------------------------------
// Phase 1: phi = relu(h @ Wh + x_t @ Wx + b);  v = alpha*h - phi
// Each wave computes a 32x16 tile (2 accumulators share B fragments).
// grid = (8 col-groups, 4 row-groups), block = 256 (8 waves)
// ---------------------------------------------------------------------------
__global__ void __launch_bounds__(256)
step_phase1(const float* __restrict__ h, const float* __restrict__ X,
            const float* __restrict__ Wh, const float* __restrict__ Wx,
            const float* __restrict__ bvec, const float* __restrict__ alpha,
            float* __restrict__ phi_out, float* __restrict__ v_out, int t) {
    const int lane = threadIdx.x & 31;
    const int wave = threadIdx.x >> 5;
    const int n    = lane & 15;
    const int half = lane >> 4;
    const int col0 = (blockIdx.x * 8 + wave) * 16;  // 64 col tiles
    const int row0 = blockIdx.y * 32;               // 4 row groups of 32

    v8f acc0 = {};
    v8f acc1 = {};

    // h @ Wh  (K = HDIM)
    #pragma unroll 4
    for (int k = 0; k < HDIM; k += 4) {
        v2f b  = load_b_frag(Wh, HDIM, k, col0, n, half);
        v2f a0 = load_a_frag(h, HDIM, row0, k, n, half);
        v2f a1 = load_a_frag(h, HDIM, row0 + 16, k, n, half);
        acc0 = __builtin_amdgcn_wmma_f32_16x16x4_f32(false, a0, false, b, (short)0, acc0, false, false);
        acc1 = __builtin_amdgcn_wmma_f32_16x16x4_f32(false, a1, false, b, (short)0, acc1, false, false);
    }
    // x_t @ Wx  (K = IDIM); X is (B, T, I), rows here are batch indices
    const float* Xt = X + (size_t)t * IDIM;
    #pragma unroll 4
    for (int k = 0; k < IDIM; k += 4) {
        v2f b  = load_b_frag(Wx, HDIM, k, col0, n, half);
        v2f a0 = load_a_frag(Xt, TDIM * IDIM, row0, k, n, half);
        v2f a1 = load_a_frag(Xt, TDIM * IDIM, row0 + 16, k, n, half);
        acc0 = __builtin_amdgcn_wmma_f32_16x16x4_f32(false, a0, false, b, (short)0, acc0, false, false);
        acc1 = __builtin_amdgcn_wmma_f32_16x16x4_f32(false, a1, false, b, (short)0, acc1, false, false);
    }

    const float a_s = alpha[0];
    const float bv  = bvec[col0 + n];
    #pragma unroll
    for (int r = 0; r < 8; ++r) {
        const int m0 = row0 + r + 8 * half;
        const int m1 = m0 + 16;
        {
            const size_t idx = (size_t)m0 * HDIM + col0 + n;
            float p = acc0[r] + bv;
            p = p > 0.f ? p : 0.f;
            phi_out[idx] = p;
            v_out[idx]   = a_s * h[idx] - p;
        }
        {
            const size_t idx = (size_t)m1 * HDIM + col0 + n;
            float p = acc1[r] + bv;
            p = p > 0.f ? p : 0.f;
            phi_out[idx] = p;
            v_out[idx]   = a_s * h[idx] - p;
        }
    }
}

// ---------------------------------------------------------------------------
// Phase 2: U = v @ Wh^T; grad = alpha*v - mask*U; h_new = h - lr*grad
// Writes h_next and outs[t].
// ---------------------------------------------------------------------------
__global__ void __launch_bounds__(256)
step_phase2(const float* __restrict__ h, const float* __restrict__ WhT,
            const float* __restrict__ phi, const float* __restrict__ v,
            const float* __restrict__ alpha, float* __restrict__ h_next,
            float* __restrict__ outs, int t) {
    const int lane = threadIdx.x & 31;
    const int wave = threadIdx.x >> 5;
    const int n    = lane & 15;
    const int half = lane >> 4;
    const int col0 = (blockIdx.x * 8 + wave) * 16;
    const int row0 = blockIdx.y * 32;

    v8f acc0 = {};
    v8f acc1 = {};

    #pragma unroll 4
    for (int k = 0; k < HDIM; k += 4) {
        v2f b  = load_b_frag(WhT, HDIM, k, col0, n, half);
        v2f a0 = load_a_frag(v, HDIM, row0, k, n, half);
        v2f a1 = load_a_frag(v, HDIM, row0 + 16, k, n, half);
        acc0 = __builtin_amdgcn_wmma_f32_16x16x4_f32(false, a0, false, b, (short)0, acc0, false, false);
        acc1 = __builtin_amdgcn_wmma_f32_16x16x4_f32(false, a1, false, b, (short)0, acc1, false, false);
    }

    const float a_s = alpha[0];
    float* outs_t = outs + (size_t)t * BDIM * HDIM;
    #pragma unroll
    for (int r = 0; r < 8; ++r) {
        const int m0 = row0 + r + 8 * half;
        const int m1 = m0 + 16;
        {
            const size_t idx = (size_t)m0 * HDIM + col0 + n;
            const float mask = phi[idx] > 0.f ? 1.f : 0.f;
            const float grad = a_s * v[idx] - mask * acc0[r];
            const float hn   = h[idx] - LRATE * grad;
            h_next[idx] = hn;
            outs_t[idx] = hn;
        }
        {
            const size_t idx = (size_t)m1 * HDIM + col0 + n;
            const float mask = phi[idx] > 0.f ? 1.f : 0.f;
            const float grad = a_s * v[idx] - mask * acc1[r];
            const float hn   = h[idx] - LRATE * grad;
            h_next[idx] = hn;
            outs_t[idx] = hn;
        }
    }
}

// ---------------------------------------------------------------------------
// One-time 1024x1024 transpose of Wh (LDS tiled)
// grid (32,32), block (32,8)
// ---------------------------------------------------------------------------
__global__ void __launch_bounds__(256)
transpose1024(const float* __restrict__ src, float* __restrict__ dst) {
    __shared__ float tile[32][33];
    int x = blockIdx.x * 32 + threadIdx.x;
    int y = blockIdx.y * 32 + threadIdx.y;
    #pragma unroll
    for (int i = 0; i < 32; i += 8)
        tile[threadIdx.y + i][threadIdx.x] = src[(size_t)(y + i) * HDIM + x];
    __syncthreads();
    x = blockIdx.y * 32 + threadIdx.x;
    y = blockIdx.x * 32 + threadIdx.y;
    #pragma unroll
    for (int i = 0; i < 32; i += 8)
        dst[(size_t)(y + i) * HDIM + x] = tile[threadIdx.x][threadIdx.y + i];
}

// ---------------------------------------------------------------------------
// Final FC: out[(b*T+t), c] = outs[t][b][:] . fc_w[:, c] + fc_b[c]
// One wave32 per output row; shuffle reduction.
// grid = B*T/8, block = 256
// ---------------------------------------------------------------------------
__global__ void __launch_bounds__(256)
fc_kernel(const float* __restrict__ outs, const float* __restrict__ fc_w,
          const float* __restrict__ fc_b, float* __restrict__ out) {
    const int lane = threadIdx.x & 31;
    const int wave = threadIdx.x >> 5;
    const int row  = blockIdx.x * 8 + wave;   // row = b*T + t
    const int b    = row / TDIM;
    const int t    = row % TDIM;
    const float* src = outs + ((size_t)t * BDIM + b) * HDIM;

    float acc[CDIM];
    #pragma unroll
    for (int c = 0; c < CDIM; ++c) acc[c] = 0.f;

    for (int hh = lane; hh < HDIM; hh += 32) {
        const float x = src[hh];
        const float* w = fc_w + (size_t)hh * CDIM;
        #pragma unroll
        for (int c = 0; c < CDIM; ++c) acc[c] += x * w[c];
    }
    #pragma unroll
    for (int c = 0; c < CDIM; ++c)
        for (int off = 16; off > 0; off >>= 1)
            acc[c] += __shfl_down(acc[c], off, 32);

    if (lane == 0) {
        float* o = out + (size_t)row * CDIM;
        #pragma unroll
        for (int c = 0; c < CDIM; ++c) o[c] = acc[c] + fc_b[c];
    }
}

// ---------------------------------------------------------------------------
extern "C" void kernel_launch(void* const* d_in, const int* in_sizes, int n_in,
                              void* d_out, int out_size, void* d_ws, size_t ws_size,
                              hipStream_t stream) {
    const float* X      = (const float*)d_in[0];  // (B,T,I)
    const float* hidden = (const float*)d_in[1];  // (B,H)
    const float* alpha  = (const float*)d_in[2];  // (1,)
    const float* bvec   = (const float*)d_in[3];  // (H,)
    const float* Wh     = (const float*)d_in[4];  // (H,H)
    const float* Wx     = (const float*)d_in[5];  // (I,H)
    const float* fc_w   = (const float*)d_in[6];  // (H,C)
    const float* fc_b   = (const float*)d_in[7];  // (C,)
    float* out = (float*)d_out;

    // workspace layout
    float* ws   = (float*)d_ws;
    float* WhT  = ws;  ws += (size_t)HDIM * HDIM;         // 4 MB
    float* h0   = ws;  ws += (size_t)BDIM * HDIM;         // 512 KB
    float* h1   = ws;  ws += (size_t)BDIM * HDIM;
    float* phi  = ws;  ws += (size_t)BDIM * HDIM;
    float* vbuf = ws;  ws += (size_t)BDIM * HDIM;
    float* outs = ws;                                     // 128 MB (T,B,H)

    hipMemcpyAsync(h0, hidden, (size_t)BDIM * HDIM * sizeof(float),
                   hipMemcpyDeviceToDevice, stream);
    transpose1024<<<dim3(32, 32), dim3(32, 8), 0, stream>>>(Wh, WhT);

    float* hb[2] = {h0, h1};
    const dim3 grid(8, 4), block(256);
    for (int t = 0; t < TDIM; ++t) {
        float* hc = hb[t & 1];
        float* hn = hb[(t + 1) & 1];
        step_phase1<<<grid, block, 0, stream>>>(hc, X, Wh, Wx, bvec, alpha, phi, vbuf, t);
        step_phase2<<<grid, block, 0, stream>>>(hc, WhT, phi, vbuf, alpha, hn, outs, t);
    }

    fc_kernel<<<(BDIM * TDIM) / 8, 256, 0, stream>>>(outs, fc_w, fc_b, out);

    // final hidden state lands in hb[0] after 256 steps
    hipMemcpyAsync(out + (size_t)BDIM * TDIM * CDIM, hb[0],
                   (size_t)BDIM * HDIM * sizeof(float),
                   hipMemcpyDeviceToDevice, stream);
}